// CustomIOU_88493506166986
// MI455X (gfx1250) — compile-verified
//
#include <hip/hip_runtime.h>

// MI455X roofline: 192 MB single-pass stream / 23.3 TB/s ≈ 8.2 us floor.
// ~180 MFLOP of VALU is negligible -> pure bandwidth kernel.
// Strategy: NT b128 (float4) coalesced grid-stride loads, f32 accumulate,
// WMMA-based wave32 reduction in the epilogue, deterministic 2-stage reduce.

typedef __attribute__((ext_vector_type(4))) float f32x4;
typedef __attribute__((ext_vector_type(2))) float v2f;
typedef __attribute__((ext_vector_type(8))) float v8f;

// Full 32-lane sum using V_WMMA_F32_16X16X4_F32.
// A layout (32-bit A 16x4): lanes 0-15 hold {K0,K1}, lanes 16-31 hold {K2,K3}.
// A = {v, 0}, B = all-ones (layout-independent since every element is 1):
//   D[m][n] = v[m] + v[m+16]  (replicated over n)
// Lane L (L<16) holds rows 0-7 in its 8 D VGPRs, lanes 16-31 hold rows 8-15.
// Sum the 8 VGPRs, then one xor-16 swap completes the 32-lane sum.
__device__ __forceinline__ float wave_sum32(float v) {
    v2f a; a[0] = v;    a[1] = 0.0f;
    v2f b; b[0] = 1.0f; b[1] = 1.0f;
    v8f c = {};
    v8f d = __builtin_amdgcn_wmma_f32_16x16x4_f32(
        /*neg_a=*/false, a, /*neg_b=*/false, b,
        /*c_mod=*/(short)0, c, /*reuse_a=*/false, /*reuse_b=*/false);
    float s = ((d[0] + d[1]) + (d[2] + d[3])) + ((d[4] + d[5]) + (d[6] + d[7]));
    s += __shfl_xor(s, 16, 32);
    return s;
}

__global__ void iou_partial_kernel(const float* __restrict__ pred,
                                   const float* __restrict__ truth,
                                   float* __restrict__ partials,
                                   int n) {
    const f32x4* __restrict__ p4 = (const f32x4*)pred;
    const f32x4* __restrict__ t4 = (const f32x4*)truth;
    float siou = 0.0f;
    float scnt = 0.0f;
    const int stride = gridDim.x * blockDim.x;
    for (int i = blockIdx.x * blockDim.x + threadIdx.x; i < n; i += stride) {
        f32x4 p = __builtin_nontemporal_load(p4 + i);  // global_load_b128, TH=NT
        f32x4 t = __builtin_nontemporal_load(t4 + i);

        float ph = 0.5f * p[2], pv = 0.5f * p[3];
        float th = 0.5f * t[2], tv = 0.5f * t[3];
        float px1 = p[0] - ph, px2 = p[0] + ph;
        float py1 = p[1] - pv, py2 = p[1] + pv;
        float tx1 = t[0] - th, tx2 = t[0] + th;
        float ty1 = t[1] - tv, ty2 = t[1] + tv;

        float x1 = fmaxf(px1, tx1), y1 = fmaxf(py1, ty1);
        float x2 = fminf(px2, tx2), y2 = fminf(py2, ty2);
        float inter = fmaxf(x2 - x1, 0.0f) * fmaxf(y2 - y1, 0.0f);
        float ap = fabsf((px2 - px1) * (py2 - py1));
        float at = fabsf((tx2 - tx1) * (ty2 - ty1));
        float iou = inter / (ap + at - inter + 1e-6f);

        // valid <=> truth row is not the sentinel [-1,-1,-1,-1]
        float vf = (t[0] == -1.0f && t[1] == -1.0f &&
                    t[2] == -1.0f && t[3] == -1.0f) ? 0.0f : 1.0f;
        siou = fmaf(iou, vf, siou);
        scnt += vf;
    }

    // All threads converged here -> EXEC all-ones, WMMA legal.
    float wiou = wave_sum32(siou);
    float wcnt = wave_sum32(scnt);

    __shared__ float sa[8];
    __shared__ float sb[8];
    const int lane = threadIdx.x & 31;
    const int wave = threadIdx.x >> 5;
    if (lane == 0) { sa[wave] = wiou; sb[wave] = wcnt; }
    __syncthreads();
    if (threadIdx.x == 0) {
        float a = 0.0f, b = 0.0f;
        #pragma unroll
        for (int w = 0; w < 8; ++w) { a += sa[w]; b += sb[w]; }
        partials[2 * blockIdx.x]     = a;
        partials[2 * blockIdx.x + 1] = b;
    }
}

__global__ void iou_final_kernel(const float* __restrict__ partials,
                                 int nblk,
                                 float* __restrict__ out) {
    float a = 0.0f, b = 0.0f;
    for (int k = threadIdx.x; k < nblk; k += blockDim.x) {
        a += partials[2 * k];
        b += partials[2 * k + 1];
    }
    float wa = wave_sum32(a);
    float wb = wave_sum32(b);

    __shared__ float sa[8];
    __shared__ float sb[8];
    const int lane = threadIdx.x & 31;
    const int wave = threadIdx.x >> 5;
    if (lane == 0) { sa[wave] = wa; sb[wave] = wb; }
    __syncthreads();
    if (threadIdx.x == 0) {
        float ta = 0.0f, tb = 0.0f;
        #pragma unroll
        for (int w = 0; w < 8; ++w) { ta += sa[w]; tb += sb[w]; }
        out[0] = (tb > 0.0f) ? (ta / fmaxf(tb, 1.0f)) : 0.0f;
    }
}

extern "C" void kernel_launch(void* const* d_in, const int* in_sizes, int n_in,
                              void* d_out, int out_size, void* d_ws, size_t ws_size,
                              hipStream_t stream) {
    const float* pred  = (const float*)d_in[0];
    const float* truth = (const float*)d_in[1];
    float* out = (float*)d_out;
    float* ws  = (float*)d_ws;   // 2048 blocks * 2 floats = 16 KB of scratch

    const int n = in_sizes[0] / 4;  // number of (cx,cy,w,h) boxes = B*NBOX
    const int BLK = 256;            // 8 wave32 per block
    const int NB  = 2048;           // 16K waves in flight; each thread ~11 iters

    hipLaunchKernelGGL(iou_partial_kernel, dim3(NB), dim3(BLK), 0, stream,
                       pred, truth, ws, n);
    hipLaunchKernelGGL(iou_final_kernel, dim3(1), dim3(BLK), 0, stream,
                       ws, NB, out);
}